// MorphologicalLayer_36575941492771
// MI455X (gfx1250) — compile-verified
//
#include <hip/hip_runtime.h>
#include <stdint.h>

// Problem sizes (fixed by the reference).
#define B_DIM   2048
#define IN_DIM  512
#define OUT_DIM 512

// Tiling: 64x64 output tile per block, K-chunks of 64, 256 threads (8 wave32).
#define BM  64
#define BN  64
#define KB  64
#define KG  (KB / 4)                 // k-groups of 4 per chunk
#define TILE_F (KG * BM * 4)         // floats per tile buffer = 4096 (16 KB)
#define NCHUNK (IN_DIM / KB)         // 8
#define NTHREADS 256

typedef __attribute__((ext_vector_type(4))) unsigned int u32x4;
typedef __attribute__((ext_vector_type(8))) int          i32x8;
typedef __attribute__((ext_vector_type(4))) int          i32x4;

struct __align__(16) f4 { float v[4]; };

#if defined(__HIP_DEVICE_COMPILE__) && __has_builtin(__builtin_amdgcn_tensor_load_to_lds)
#define USE_TDM 1
#else
#define USE_TDM 0
#endif

#if USE_TDM
// 3D TDM tile: x = 4 contiguous k (quad), y = 64 rows (stride IN_DIM),
// z = KG k-groups (stride 4).  LDS receives [z][y][x] = [kgroup][row][kq],
// which makes every 4-k quad of a row a 16B-aligned float4 in LDS.
__device__ __forceinline__ i32x8 tdm_make_group1() {
  i32x8 g1;
  g1[0] = (int)(2u << 16);                         // data_size=2 (4B); no pad; mask=0
  g1[1] = (int)((uint32_t)IN_DIM << 16);           // tensor_dim0 lo16 at bits[63:48]
  g1[2] = (int)((uint32_t)B_DIM  << 16);           // dim0 hi=0; tensor_dim1 lo16 at [95:80]
  g1[3] = (int)(4u << 16);                         // dim1 hi=0; tile_dim0=4 at [127:112]
  g1[4] = (int)((uint32_t)BM | ((uint32_t)KG << 16)); // tile_dim1=64, tile_dim2=KG
  g1[5] = IN_DIM;                                  // tensor_dim0_stride lo32
  g1[6] = (int)(4u << 16);                         // stride0 hi=0; tensor_dim1_stride lo16=4
  g1[7] = 0;                                       // stride1 hi
  return g1;
}

__device__ __forceinline__ void tdm_load_tile(uint32_t lds_addr, uint64_t gaddr,
                                              const i32x8& g1) {
  u32x4 g0;
  g0[0] = 1u;                                      // count=1, user descriptor, no gather
  g0[1] = lds_addr;                                // LDS byte address
  g0[2] = (uint32_t)gaddr;                         // global_addr[31:0]
  g0[3] = ((uint32_t)(gaddr >> 32) & 0x01FFFFFFu) | (2u << 30); // addr[56:32] | type=2
  i32x4 g2 = {IN_DIM / 4, 0, 0, 0};                // tensor_dim2 = total k-groups
  i32x4 g3 = {0, 0, 0, 0};
  i32x8 z8 = {0, 0, 0, 0, 0, 0, 0, 0};
  __builtin_amdgcn_tensor_load_to_lds(g0, g1, g2, g3, z8, 0);
}

__device__ __forceinline__ uint32_t lds_addr_of(const void* p) {
  // Generic LDS pointer: aperture in high 32 bits, LDS byte offset in low 32.
  return (uint32_t)(uintptr_t)p;
}
#endif

// 4x4 register-tile tropical update over one KB-wide chunk in LDS.
// Layout: f4 tile[KG][64] ; quad q of row r lives at tile[q][r].
__device__ __forceinline__ void compute_chunk(const float* __restrict__ xbase,
                                              const float* __restrict__ dbase,
                                              const float* __restrict__ ebase,
                                              int ty, int tx,
                                              float (&dil)[4][4], float (&ero)[4][4]) {
  const f4* __restrict__ xp = (const f4*)xbase;
  const f4* __restrict__ dp = (const f4*)dbase;
  const f4* __restrict__ ep = (const f4*)ebase;
#pragma unroll 2
  for (int kk = 0; kk < KG; ++kk) {
    f4 a[4], bd[4], be[4];
#pragma unroll
    for (int i = 0; i < 4; ++i) a[i] = xp[kk * BM + 4 * ty + i];      // wave-broadcast
#pragma unroll
    for (int j = 0; j < 4; ++j) {
      bd[j] = dp[kk * BN + 4 * tx + j];                               // 64-bank sweep
      be[j] = ep[kk * BN + 4 * tx + j];
    }
#pragma unroll
    for (int kq = 0; kq < 4; ++kq) {
#pragma unroll
      for (int i = 0; i < 4; ++i) {
#pragma unroll
        for (int j = 0; j < 4; ++j) {
          dil[i][j] = fmaxf(dil[i][j], a[i].v[kq] + bd[j].v[kq]);
          ero[i][j] = fminf(ero[i][j], a[i].v[kq] - be[j].v[kq]);
        }
      }
    }
  }
}

__global__ __launch_bounds__(NTHREADS)
void morph_tropical_kernel(const float* __restrict__ x,
                           const float* __restrict__ wd,
                           const float* __restrict__ we,
                           float* __restrict__ out) {
  // Double-buffered LDS tiles in [kgroup][row][kq] interleave (16 KB each).
  __shared__ __align__(16) float xs[2][TILE_F];
  __shared__ __align__(16) float ds[2][TILE_F];
  __shared__ __align__(16) float es[2][TILE_F];

  const int tid = threadIdx.x;
  const int tx = tid & 15;        // output-column group
  const int ty = tid >> 4;        // batch-row group
  const int bn0 = blockIdx.x * BN;
  const int bm0 = blockIdx.y * BM;

  float dil[4][4], ero[4][4];
#pragma unroll
  for (int i = 0; i < 4; ++i)
#pragma unroll
    for (int j = 0; j < 4; ++j) {
      dil[i][j] = -__builtin_inff();
      ero[i][j] =  __builtin_inff();
    }

#if USE_TDM
  const uint64_t xg = (uint64_t)(uintptr_t)x  + 4ull * ((uint64_t)bm0 * IN_DIM);
  const uint64_t dg = (uint64_t)(uintptr_t)wd + 4ull * ((uint64_t)bn0 * IN_DIM);
  const uint64_t eg = (uint64_t)(uintptr_t)we + 4ull * ((uint64_t)bn0 * IN_DIM);
  const i32x8 g1 = tdm_make_group1();
  // Scalar (SGPR) wave-id gate: only wave 0 issues TDM / waits TENSORcnt.
  const bool issuer = (__builtin_amdgcn_readfirstlane(tid) == 0);

  if (issuer) {                               // prologue: chunk 0 -> buffer 0
    tdm_load_tile(lds_addr_of(&xs[0][0]), xg, g1);
    tdm_load_tile(lds_addr_of(&ds[0][0]), dg, g1);
    tdm_load_tile(lds_addr_of(&es[0][0]), eg, g1);
  }

#pragma unroll 1
  for (int c = 0; c < NCHUNK; ++c) {
    const int cur = c & 1;
    if (issuer) {
      if (c + 1 < NCHUNK) {                   // prefetch next chunk into other buffer
        const uint64_t off = 4ull * (uint64_t)((c + 1) * KB);
        tdm_load_tile(lds_addr_of(&xs[cur ^ 1][0]), xg + off, g1);
        tdm_load_tile(lds_addr_of(&ds[cur ^ 1][0]), dg + off, g1);
        tdm_load_tile(lds_addr_of(&es[cur ^ 1][0]), eg + off, g1);
        __builtin_amdgcn_s_wait_tensorcnt(3); // current chunk's 3 loads complete
      } else {
        __builtin_amdgcn_s_wait_tensorcnt(0);
      }
    }
    __syncthreads();
    compute_chunk(&xs[cur][0], &ds[cur][0], &es[cur][0], ty, tx, dil, ero);
    __syncthreads();                          // buf(cur) reusable two chunks later
  }
#else
  // Fallback (host pass / missing builtin): cooperative loads, same LDS layout.
#pragma unroll 1
  for (int c = 0; c < NCHUNK; ++c) {
    const int cur = c & 1;
    for (int idx = tid; idx < BM * KB; idx += NTHREADS) {
      const int r = idx / KB, k = idx % KB;
      const int li = ((k >> 2) * BM + r) * 4 + (k & 3);
      xs[cur][li] = x[(size_t)(bm0 + r) * IN_DIM + c * KB + k];
    }
    for (int idx = tid; idx < BN * KB; idx += NTHREADS) {
      const int r = idx / KB, k = idx % KB;
      const int li = ((k >> 2) * BN + r) * 4 + (k & 3);
      ds[cur][li] = wd[(size_t)(bn0 + r) * IN_DIM + c * KB + k];
      es[cur][li] = we[(size_t)(bn0 + r) * IN_DIM + c * KB + k];
    }
    __syncthreads();
    compute_chunk(&xs[cur][0], &ds[cur][0], &es[cur][0], ty, tx, dil, ero);
    __syncthreads();
  }
#endif

  // Epilogue: out[b, o] = dilation + erosion; 4 consecutive columns -> float4 stores.
#pragma unroll
  for (int i = 0; i < 4; ++i) {
    const int row = bm0 + 4 * ty + i;
    float4 v;
    v.x = dil[i][0] + ero[i][0];
    v.y = dil[i][1] + ero[i][1];
    v.z = dil[i][2] + ero[i][2];
    v.w = dil[i][3] + ero[i][3];
    *(float4*)&out[(size_t)row * OUT_DIM + bn0 + 4 * tx] = v;
  }
}

extern "C" void kernel_launch(void* const* d_in, const int* in_sizes, int n_in,
                              void* d_out, int out_size, void* d_ws, size_t ws_size,
                              hipStream_t stream) {
  (void)in_sizes; (void)n_in; (void)out_size; (void)d_ws; (void)ws_size;
  const float* x  = (const float*)d_in[0];
  const float* wd = (const float*)d_in[1];
  const float* we = (const float*)d_in[2];
  float* out = (float*)d_out;

  dim3 grid(OUT_DIM / BN, B_DIM / BM);   // 8 x 32 = 256 workgroups
  dim3 block(NTHREADS);
  morph_tropical_kernel<<<grid, block, 0, stream>>>(x, wd, we, out);
}